// AttentionBasedExperts_29386166239541
// MI455X (gfx1250) — compile-verified
//
#include <hip/hip_runtime.h>

typedef __attribute__((ext_vector_type(2))) float v2f;
typedef __attribute__((ext_vector_type(8))) float v8f;

#define NUM_TASKS   50
#define EMB_DIM     128
#define HIDDEN      256
#define NUM_EXPERTS 16
#define MPAD        64   // 50 tasks padded to 4 M-tiles of 16

// ---------------------------------------------------------------------------
// One wave (32 threads) computes one 16x16 tile of  Out = act(A @ B + bias)
// using V_WMMA_F32_16X16X4_F32 with a K-loop of step 4.
//
// fp32 WMMA VGPR layouts (ISA 7.12.2):
//   A (16x4):  lane L -> row m = L%16;  lanes 0-15 hold K=k0,k0+1,
//              lanes 16-31 hold K=k0+2,k0+3  (2 VGPRs = v2f)
//   B (4x16):  lane L -> col n = L%16;  same K split as A
//   C/D:       acc[i] = D[M = 16*tm + 8*(L/16) + i][N = 16*tn + L%16]
// ---------------------------------------------------------------------------
template<bool CLAMP_ROWS, bool RELU>
__global__ void __launch_bounds__(32)
gemm_f32_wmma(const float* __restrict__ A, int lda, int arows,
              const float* __restrict__ Bw, int ldb,
              const float* __restrict__ bias,
              float* __restrict__ Out, int ldo,
              int K, int ntn)
{
    const int tile = blockIdx.x;
    const int tm   = tile / ntn;
    const int tn   = tile - tm * ntn;
    const int lane = threadIdx.x;       // 0..31, full wave (EXEC all ones)
    const int lo   = lane & 15;
    const int hi   = lane >> 4;         // K-pair selector
    const int m    = tm * 16 + lo;
    const int n    = tn * 16 + lo;

    // Zero-pad rows >= arows without reading out of bounds (and without
    // EXEC divergence around the WMMA).
    int   mr = m;
    float ms = 1.0f;
    if (CLAMP_ROWS && m >= arows) { mr = arows - 1; ms = 0.0f; }

    const float* Arow = A  + (size_t)mr * lda + hi * 2;   // A[mr][k0 + hi*2 + ...]
    const float* Bcol = Bw + (size_t)(hi * 2) * ldb + n;  // B[k0 + hi*2 + ...][n]

    v8f acc = {};
    #pragma unroll 4
    for (int k0 = 0; k0 < K; k0 += 4) {
        v2f a, b;
        a.x = Arow[k0]     * ms;
        a.y = Arow[k0 + 1] * ms;
        b.x = Bcol[(size_t)k0 * ldb];
        b.y = Bcol[(size_t)k0 * ldb + ldb];
        // 8 args: (neg_a, A, neg_b, B, c_mod, C, reuse_a, reuse_b)
        acc = __builtin_amdgcn_wmma_f32_16x16x4_f32(false, a, false, b,
                                                    (short)0, acc, false, false);
    }

    const float bv = bias[n];
    float* orow = Out + (size_t)(tm * 16 + hi * 8) * ldo + n;
    #pragma unroll
    for (int i = 0; i < 8; ++i) {
        float v = acc[i] + bv;
        if (RELU) v = v > 0.0f ? v : 0.0f;
        orow[(size_t)i * ldo] = v;
    }
}

// ---------------------------------------------------------------------------
// Per-task softmax + hard top-2 + renormalize.  50 tasks -> 50 threads.
// ---------------------------------------------------------------------------
__global__ void __launch_bounds__(64)
gate_softmax_top2(const float* __restrict__ logits,   // [MPAD][NUM_EXPERTS]
                  float* __restrict__ gate)           // [NUM_TASKS][NUM_EXPERTS]
{
    const int t = blockIdx.x * blockDim.x + threadIdx.x;
    if (t >= NUM_TASKS) return;

    float l[NUM_EXPERTS];
    float mx = -3.4e38f;
    #pragma unroll
    for (int e = 0; e < NUM_EXPERTS; ++e) {
        l[e] = logits[t * NUM_EXPERTS + e];   // TEMPERATURE == 1.0
        mx = l[e] > mx ? l[e] : mx;
    }
    float sum = 0.0f;
    float p[NUM_EXPERTS];
    #pragma unroll
    for (int e = 0; e < NUM_EXPERTS; ++e) {
        p[e] = __expf(l[e] - mx);
        sum += p[e];
    }
    const float inv = 1.0f / sum;
    #pragma unroll
    for (int e = 0; e < NUM_EXPERTS; ++e) p[e] *= inv;

    // top-2 (first occurrence wins ties, matching lax.top_k)
    int i1 = 0; float v1 = p[0];
    #pragma unroll
    for (int e = 1; e < NUM_EXPERTS; ++e)
        if (p[e] > v1) { v1 = p[e]; i1 = e; }
    int i2 = -1; float v2 = -1.0f;
    #pragma unroll
    for (int e = 0; e < NUM_EXPERTS; ++e)
        if (e != i1 && p[e] > v2) { v2 = p[e]; i2 = e; }

    const float rs = 1.0f / (v1 + v2);
    #pragma unroll
    for (int e = 0; e < NUM_EXPERTS; ++e)
        gate[t * NUM_EXPERTS + e] = (e == i1 || e == i2) ? p[e] * rs : 0.0f;
}

// ---------------------------------------------------------------------------
// Scatter: out[e*B + b] = gate[env[b]][e].  Bandwidth-bound phase:
// 8 MB coalesced stores + 0.5 MB index reads.  Gate table cached in LDS.
// ---------------------------------------------------------------------------
__global__ void __launch_bounds__(256)
gate_scatter(const int* __restrict__ env,
             const float* __restrict__ gate,
             float* __restrict__ out, int Bn)
{
    __shared__ float g[NUM_TASKS * NUM_EXPERTS];   // 3.2 KB
    for (int i = threadIdx.x; i < NUM_TASKS * NUM_EXPERTS; i += blockDim.x)
        g[i] = gate[i];
    __syncthreads();

    const int b = blockIdx.x * blockDim.x + threadIdx.x;
    if (b >= Bn) return;
    const int   t   = env[b];
    const float* gr = &g[t * NUM_EXPERTS];
    #pragma unroll
    for (int e = 0; e < NUM_EXPERTS; ++e)   // consecutive lanes -> consecutive b:
        out[(size_t)e * Bn + b] = gr[e];    // every store fully coalesced
}

// ---------------------------------------------------------------------------
extern "C" void kernel_launch(void* const* d_in, const int* in_sizes, int n_in,
                              void* d_out, int out_size, void* d_ws, size_t ws_size,
                              hipStream_t stream)
{
    const int*   env = (const int*)  d_in[0];
    const float* emb = (const float*)d_in[1];   // [50][128]
    const float* W1  = (const float*)d_in[2];   // [128][256]
    const float* b1  = (const float*)d_in[3];
    const float* W2  = (const float*)d_in[4];   // [256][256]
    const float* b2  = (const float*)d_in[5];
    const float* W3  = (const float*)d_in[6];   // [256][16]
    const float* b3  = (const float*)d_in[7];
    float* out = (float*)d_out;
    const int Bn = in_sizes[0];                 // 131072

    float* ws     = (float*)d_ws;
    float* h1     = ws;                              // [64][256]
    float* h2     = h1 + MPAD * HIDDEN;              // [64][256]
    float* logits = h2 + MPAD * HIDDEN;              // [64][16]
    float* gate   = logits + MPAD * NUM_EXPERTS;     // [50][16]

    // Layer 1: h1 = relu(emb @ W1 + b1)  — rows >= 50 zero-padded
    gemm_f32_wmma<true,  true ><<<4 * (HIDDEN / 16), 32, 0, stream>>>(
        emb, EMB_DIM, NUM_TASKS, W1, HIDDEN, b1, h1, HIDDEN, EMB_DIM, HIDDEN / 16);
    // Layer 2: h2 = relu(h1 @ W2 + b2)
    gemm_f32_wmma<false, true ><<<4 * (HIDDEN / 16), 32, 0, stream>>>(
        h1, HIDDEN, MPAD, W2, HIDDEN, b2, h2, HIDDEN, HIDDEN, HIDDEN / 16);
    // Layer 3: logits = h2 @ W3 + b3
    gemm_f32_wmma<false, false><<<4 * (NUM_EXPERTS / 16), 32, 0, stream>>>(
        h2, HIDDEN, MPAD, W3, NUM_EXPERTS, b3, logits, NUM_EXPERTS, HIDDEN,
        NUM_EXPERTS / 16);

    // softmax + top-2 + renormalize for the 50 tasks
    gate_softmax_top2<<<1, 64, 0, stream>>>(logits, gate);

    // bandwidth phase: broadcast per-task gates to all B tokens
    gate_scatter<<<(Bn + 255) / 256, 256, 0, stream>>>(env, gate, out, Bn);
}